// NNGLS_19456201851269
// MI455X (gfx1250) — compile-verified
//
#include <hip/hip_runtime.h>

typedef float v2f __attribute__((ext_vector_type(2)));
typedef float v8f __attribute__((ext_vector_type(8)));

constexpr int F_INC = 16;   // input features
constexpr int HC    = 64;   // hidden width
constexpr int KNC   = 20;   // neighbors per node

static __device__ __forceinline__ float lane_bcast(float v, int srcLane) {
    // constant-lane broadcast -> v_readlane_b32 (SGPR), valid: all 32 lanes active
    return __int_as_float(__builtin_amdgcn_readlane(__float_as_int(v), srcLane));
}

// ---------------------------------------------------------------------------
// Kernel 1: MLP  o = relu(x @ w1 + b1) @ w2 + b2   via V_WMMA_F32_16X16X4_F32
// One wave handles a 16-node row tile; 4 col tiles x 4 chained K=4 WMMAs.
// ---------------------------------------------------------------------------
__global__ void nngls_mlp_wmma(const float* __restrict__ x,
                               const float* __restrict__ w1,
                               const float* __restrict__ b1,
                               const float* __restrict__ w2,
                               const float* __restrict__ b2,
                               float* __restrict__ o, int N) {
    __shared__ float hbuf[4][16][HC + 1];
    const int tid  = threadIdx.x;
    const int wave = tid >> 5;
    const int lane = tid & 31;
    const int hlf  = lane >> 4;      // which 16-lane half
    const int nn   = lane & 15;
    const int base = (blockIdx.x * 4 + wave) * 16;

    // A fragment: lane holds row m = nn, K pair at kb = 2*hlf (per ISA 16x4 f32 layout)
    const int gm = base + nn;
    const int cm = gm < N ? gm : N - 1;        // clamp: keep EXEC all-ones for WMMA
    const int kb = hlf * 2;
    v2f a[4];
#pragma unroll
    for (int s = 0; s < 4; ++s) {
        a[s].x = x[cm * F_INC + 4 * s + kb];
        a[s].y = x[cm * F_INC + 4 * s + kb + 1];
    }

#pragma unroll
    for (int t = 0; t < 4; ++t) {              // 4 column tiles of 16 (H=64)
        const float bb = b1[16 * t + nn];
        v8f acc;
#pragma unroll
        for (int j = 0; j < 8; ++j) acc[j] = bb;   // C init = bias (same for all rows)
#pragma unroll
        for (int s = 0; s < 4; ++s) {          // chain K: 4 x (16x16x4)
            v2f bf;                            // B fragment: row kb(+1), col nn
            bf.x = w1[(4 * s + kb) * HC + 16 * t + nn];
            bf.y = w1[(4 * s + kb + 1) * HC + 16 * t + nn];
            acc = __builtin_amdgcn_wmma_f32_16x16x4_f32(
                false, a[s], false, bf, (short)0, acc, false, false);
        }
        // ReLU + park tile in LDS (C layout: VGPR j -> row j + 8*hlf, col 16t+nn)
#pragma unroll
        for (int j = 0; j < 8; ++j) {
            float v = acc[j] > 0.0f ? acc[j] : 0.0f;
            hbuf[wave][j + 8 * hlf][16 * t + nn] = v;
        }
    }
    __syncthreads();

    // Second layer: 64 -> 1 dot per node
    if (lane < 16) {
        const int node = base + lane;
        if (node < N) {
            float accv = b2[0];
#pragma unroll
            for (int c = 0; c < HC; ++c) accv += hbuf[wave][lane][c] * w2[c];
            o[node] = accv;
        }
    }
}

// ---------------------------------------------------------------------------
// Kernel 2: scatter edge source index into dense [N,K] table.
// (dst,nbr) pairs are unique by construction -> race-free, deterministic.
// ---------------------------------------------------------------------------
__global__ void nngls_scatter_src(const int* __restrict__ ei,
                                  const int* __restrict__ ea,
                                  int* __restrict__ ws_src, int E, int K) {
    const int e = blockIdx.x * blockDim.x + threadIdx.x;
    if (e >= E) return;
    const int s = ei[e];
    const int d = ei[E + e];
    const int k = ea[e];
    ws_src[d * K + k] = s;
}

// ---------------------------------------------------------------------------
// Kernel 3: per-node 20x20 SPD solve, one node per wave32.
// Lane r owns row r; pivot/broadcasts via v_readlane (constant lanes).
// ---------------------------------------------------------------------------
__global__ void nngls_solve(const float* __restrict__ pos,
                            const float* __restrict__ y,
                            const float* __restrict__ o,
                            const int* __restrict__ ws_src,
                            const float* __restrict__ sigma,
                            const float* __restrict__ phi,
                            const float* __restrict__ tau,
                            float* __restrict__ out, int N) {
    const int wave = threadIdx.x >> 5;
    const int lane = threadIdx.x & 31;
    const int i = blockIdx.x * (blockDim.x >> 5) + wave;
    if (i >= N) return;   // uniform per wave

    const float s2 = sigma[0] * sigma[0];
    const float ph = phi[0];
    const float t2 = tau[0] * tau[0];

    const bool valid = lane < KNC;
    const int  sIdx  = valid ? ws_src[i * KNC + lane] : i;
    const float px = pos[2 * sIdx], py = pos[2 * sIdx + 1];
    const float yk = y[sIdx];
    const float ok = o[sIdx];
    const float pix = pos[2 * i], piy = pos[2 * i + 1];

    // cov_vec entry for slot `lane`
    const float dx = pix - px, dy = piy - py;
    const float cov = valid ? s2 * __expf(-sqrtf(dx * dx + dy * dy) / ph) : 0.0f;

    // Build row `lane` of C = s2*exp(-d/ph) + t2*I  (neighbor-neighbor)
    float Crow[KNC];
#pragma unroll
    for (int c = 0; c < KNC; ++c) {
        const float pcx = lane_bcast(px, c);
        const float pcy = lane_bcast(py, c);
        const float ex = px - pcx, ey = py - pcy;
        float v = s2 * __expf(-sqrtf(ex * ex + ey * ey) / ph);
        if (c == lane) v += t2;
        Crow[c] = v;
    }
    float rhs = cov;

    // Forward elimination (SPD -> no pivoting). Lane p's row is the pivot row.
#pragma unroll
    for (int p = 0; p < KNC; ++p) {
        const float piv = lane_bcast(Crow[p], p);
        const float f = (lane > p) ? Crow[p] / piv : 0.0f;
#pragma unroll
        for (int c = p; c < KNC; ++c) Crow[c] -= f * lane_bcast(Crow[c], p);
        rhs -= f * lane_bcast(rhs, p);
    }

    // Back substitution; xv[] becomes wave-uniform (readlane) solution B = C^-1 cov
    float xv[KNC];
#pragma unroll
    for (int p = KNC - 1; p >= 0; --p) {
        float acc = rhs;
#pragma unroll
        for (int c = p + 1; c < KNC; ++c) acc -= Crow[c] * xv[c];
        const float xp = acc / Crow[p];     // meaningful in lane p
        xv[p] = lane_bcast(xp, p);
    }

    // B.cov, B.y_nbr, B.o_nbr via constant-lane broadcasts
    float bc = 0.0f, by = 0.0f, bo = 0.0f;
#pragma unroll
    for (int c = 0; c < KNC; ++c) {
        bc += xv[c] * lane_bcast(cov, c);
        by += xv[c] * lane_bcast(yk, c);
        bo += xv[c] * lane_bcast(ok, c);
    }

    if (lane == 0) {
        const float Fv  = s2 + t2 - bc;
        const float isf = 1.0f / sqrtf(Fv);
        const float yi = y[i];
        const float oi = o[i];
        out[3 * i + 0] = (yi - by) * isf;
        out[3 * i + 1] = (oi - bo) * isf;
        out[3 * i + 2] = oi;
    }
}

// ---------------------------------------------------------------------------
extern "C" void kernel_launch(void* const* d_in, const int* in_sizes, int n_in,
                              void* d_out, int out_size, void* d_ws, size_t ws_size,
                              hipStream_t stream) {
    const float* x     = (const float*)d_in[0];
    const float* y     = (const float*)d_in[1];
    const float* pos   = (const float*)d_in[2];
    const float* w1    = (const float*)d_in[3];
    const float* b1    = (const float*)d_in[4];
    const float* w2    = (const float*)d_in[5];
    const float* b2    = (const float*)d_in[6];
    const float* sigma = (const float*)d_in[7];
    const float* phi   = (const float*)d_in[8];
    const float* tau   = (const float*)d_in[9];
    const int*   eidx  = (const int*)d_in[10];
    const int*   eattr = (const int*)d_in[11];

    const int N = in_sizes[1];     // y is [N]
    const int E = in_sizes[11];    // edge_attr is [E] = N*K

    int*   ws_src = (int*)d_ws;
    float* ws_o   = (float*)((char*)d_ws + (size_t)E * sizeof(int));
    float* out    = (float*)d_out;

    // 1) MLP (WMMA): 64 nodes / 128-thread block (4 waves x 16 nodes)
    const int mlpBlocks = (N + 63) / 64;
    nngls_mlp_wmma<<<mlpBlocks, 128, 0, stream>>>(x, w1, b1, w2, b2, ws_o, N);

    // 2) Edge scatter of source indices
    nngls_scatter_src<<<(E + 255) / 256, 256, 0, stream>>>(eidx, eattr, ws_src, E, KNC);

    // 3) Per-node solve: one node per wave, 8 waves / block
    nngls_solve<<<(N + 7) / 8, 256, 0, stream>>>(pos, y, ws_o, ws_src,
                                                 sigma, phi, tau, out, N);
}